// GCN_44641890074931
// MI455X (gfx1250) — compile-verified
//
#include <hip/hip_runtime.h>
#include <math.h>

typedef float v2f __attribute__((ext_vector_type(2)));
typedef float v8f __attribute__((ext_vector_type(8)));

#define FEAT 128
#define NCLASS 40
#define NCLASS_PAD 48
#define TB 256

// ---------------- utility kernels ----------------

__global__ void zero_i32(int* __restrict__ p, int n) {
    int i = blockIdx.x * blockDim.x + threadIdx.x;
    if (i < n) p[i] = 0;
}

__global__ void count_edges(const int* __restrict__ col, int* __restrict__ cnt, int E) {
    int i = blockIdx.x * blockDim.x + threadIdx.x;
    if (i < E) atomicAdd(&cnt[col[i]], 1);
}

// dinv[i] = 1/sqrt(in_degree + 1)   (self-loop included, matches reference)
__global__ void make_dinv(float* __restrict__ dinv, const int* __restrict__ cnt, int n) {
    int i = blockIdx.x * blockDim.x + threadIdx.x;
    if (i < n) dinv[i] = rsqrtf((float)cnt[i] + 1.0f);
}

// pad W3 [128 x 40] -> [128 x 48] zero-filled
__global__ void pad_w3(const float* __restrict__ W3, float* __restrict__ W3p) {
    int i = blockIdx.x * blockDim.x + threadIdx.x;
    if (i < FEAT * NCLASS_PAD) {
        int r = i / NCLASS_PAD, c = i - r * NCLASS_PAD;
        W3p[i] = (c < NCLASS) ? W3[r * NCLASS + c] : 0.0f;
    }
}

// ---------------- exclusive scan (3-phase, n <= 65536) ----------------

__global__ void scan_block(const int* __restrict__ cnt, int* __restrict__ off,
                           int* __restrict__ bsums, int n) {
    __shared__ int sh[TB];
    int i = blockIdx.x * TB + threadIdx.x;
    int v = (i < n) ? cnt[i] : 0;
    sh[threadIdx.x] = v;
    __syncthreads();
#pragma unroll
    for (int d = 1; d < TB; d <<= 1) {
        int t = (threadIdx.x >= d) ? sh[threadIdx.x - d] : 0;
        __syncthreads();
        sh[threadIdx.x] += t;
        __syncthreads();
    }
    if (i < n) off[i + 1] = sh[threadIdx.x];           // local inclusive
    if (threadIdx.x == TB - 1) bsums[blockIdx.x] = sh[TB - 1];
}

// single block: exclusive scan of per-block sums (nb <= 256)
__global__ void scan_sums(int* __restrict__ bsums, int nb) {
    __shared__ int sh[TB];
    int v = (threadIdx.x < nb) ? bsums[threadIdx.x] : 0;
    sh[threadIdx.x] = v;
    __syncthreads();
#pragma unroll
    for (int d = 1; d < TB; d <<= 1) {
        int t = (threadIdx.x >= d) ? sh[threadIdx.x - d] : 0;
        __syncthreads();
        sh[threadIdx.x] += t;
        __syncthreads();
    }
    if (threadIdx.x < nb) bsums[threadIdx.x] = sh[threadIdx.x] - v;  // exclusive
}

__global__ void scan_add(int* __restrict__ off, const int* __restrict__ bsums, int n) {
    int i = blockIdx.x * TB + threadIdx.x;
    if (i == 0) off[0] = 0;
    if (i < n) off[i + 1] += bsums[blockIdx.x];
}

// CSR fill: slot per incoming edge of each destination node
__global__ void csr_fill(const int* __restrict__ row, const int* __restrict__ col,
                         const int* __restrict__ off, int* __restrict__ cursor,
                         const float* __restrict__ dinv, int* __restrict__ srcIdx,
                         float* __restrict__ normv, int E) {
    int e = blockIdx.x * blockDim.x + threadIdx.x;
    if (e >= E) return;
    int r = row[e], c = col[e];
    int p = off[c] + atomicAdd(&cursor[c], 1);
    srcIdx[p] = r;
    normv[p] = dinv[r] * dinv[c];
}

// ---------------- WMMA fp32 GEMM ----------------
// C[M x N] = A[M x 128] * B[128 x N], N == ldb == ldc, M % 16 == 0.
// blockDim.x = (N/16)*32; each wave computes one 16x16 tile.
// V_WMMA_F32_16X16X4_F32 layouts (ISA 7.12.2):
//   A 16x4: lanes 0-15 -> M=lane, v0=K0,v1=K1; lanes 16-31 -> v0=K2,v1=K3
//   B 4x16: lanes 0-15 -> N=lane, v0=K0,v1=K1; lanes 16-31 -> v0=K2,v1=K3
//   C/D   : VGPR i -> M = i + 8*(lane>=16), N = lane&15
__global__ void gemm_wmma_f32(const float* __restrict__ A, const float* __restrict__ B,
                              float* __restrict__ C, int N) {
    const int wave     = threadIdx.x >> 5;
    const int lane     = threadIdx.x & 31;
    const int laneIdx  = lane & 15;
    const int laneHalf = lane >> 4;
    const int row0     = blockIdx.x * 16;
    const int col0     = wave * 16;

    const float* Arow = A + (size_t)(row0 + laneIdx) * FEAT + laneHalf * 2;
    const float* Bcol = B + (size_t)(laneHalf * 2) * N + col0 + laneIdx;

    v8f acc = {};
#pragma unroll 8
    for (int k = 0; k < FEAT; k += 4) {
        v2f a, b;
        a.x = Arow[k];
        a.y = Arow[k + 1];
        b.x = Bcol[(size_t)k * N];
        b.y = Bcol[(size_t)k * N + N];
        acc = __builtin_amdgcn_wmma_f32_16x16x4_f32(
            /*neg_a=*/false, a, /*neg_b=*/false, b,
            /*c_mod=*/(short)0, acc, /*reuse_a=*/false, /*reuse_b=*/false);
    }

    float* Crow = C + (size_t)row0 * N + col0 + laneIdx;
#pragma unroll
    for (int i = 0; i < 8; ++i) {
        Crow[(size_t)(i + 8 * laneHalf) * N] = acc[i];
    }
}

// ---------------- aggregation (atomic-free gather) ----------------

// One wave per node, float4 per lane (128 features).
// out[node] = relu( dinv^2 * xw[node] + bias + sum_in normv * xw[src] )
template <bool RELU>
__global__ void gather128(float* __restrict__ out, const float* __restrict__ xw,
                          const float* __restrict__ dinv, const float* __restrict__ bias,
                          const int* __restrict__ off, const int* __restrict__ srcIdx,
                          const float* __restrict__ normv, int n) {
    int node = (blockIdx.x * blockDim.x + threadIdx.x) >> 5;
    int lane = threadIdx.x & 31;
    if (node >= n) return;

    float d  = dinv[node];
    float dd = d * d;
    const float4 self = *reinterpret_cast<const float4*>(xw + (size_t)node * FEAT + lane * 4);
    const float4 bb   = *reinterpret_cast<const float4*>(bias + lane * 4);
    float4 acc;
    acc.x = dd * self.x + bb.x;
    acc.y = dd * self.y + bb.y;
    acc.z = dd * self.z + bb.z;
    acc.w = dd * self.w + bb.w;

    int s = off[node], t = off[node + 1];
    for (int p = s; p < t; p += 32) {
        int rem  = t - p;
        int idx  = (lane < rem) ? srcIdx[p + lane] : 0;   // coalesced chunk load
        float w  = (lane < rem) ? normv[p + lane] : 0.0f;
        int cnt  = rem < 32 ? rem : 32;
        for (int j = 0; j < cnt; ++j) {
            int   srcn = __shfl(idx, j, 32);
            float wj   = __shfl(w, j, 32);
            const float4 v = *reinterpret_cast<const float4*>(xw + (size_t)srcn * FEAT + lane * 4);
            acc.x += wj * v.x;
            acc.y += wj * v.y;
            acc.z += wj * v.z;
            acc.w += wj * v.w;
        }
    }
    if (RELU) {
        acc.x = fmaxf(acc.x, 0.0f);
        acc.y = fmaxf(acc.y, 0.0f);
        acc.z = fmaxf(acc.z, 0.0f);
        acc.w = fmaxf(acc.w, 0.0f);
    }
    *reinterpret_cast<float4*>(out + (size_t)node * FEAT + lane * 4) = acc;
}

// Layer 3: gather 40 features (xw leading dim 48) + bias, then fused log_softmax.
// One wave per node; lane holds f=lane, lanes 0..7 also hold f=lane+32.
__global__ void gather40_lsm(float* __restrict__ out, const float* __restrict__ xw,
                             const float* __restrict__ dinv, const float* __restrict__ bias,
                             const int* __restrict__ off, const int* __restrict__ srcIdx,
                             const float* __restrict__ normv, int n) {
    int node = (blockIdx.x * blockDim.x + threadIdx.x) >> 5;
    int lane = threadIdx.x & 31;
    if (node >= n) return;

    float d  = dinv[node];
    float dd = d * d;
    bool  hi = lane < (NCLASS - 32);
    const float* selfp = xw + (size_t)node * NCLASS_PAD;
    float v0 = dd * selfp[lane] + bias[lane];
    float v1 = hi ? (dd * selfp[lane + 32] + bias[lane + 32]) : 0.0f;

    int s = off[node], t = off[node + 1];
    for (int p = s; p < t; p += 32) {
        int rem  = t - p;
        int idx  = (lane < rem) ? srcIdx[p + lane] : 0;
        float w  = (lane < rem) ? normv[p + lane] : 0.0f;
        int cnt  = rem < 32 ? rem : 32;
        for (int j = 0; j < cnt; ++j) {
            int   srcn = __shfl(idx, j, 32);
            float wj   = __shfl(w, j, 32);
            const float* sp = xw + (size_t)srcn * NCLASS_PAD;
            v0 += wj * sp[lane];
            if (hi) v1 += wj * sp[lane + 32];
        }
    }

    // log_softmax over the 40 values held by this wave
    float m = hi ? fmaxf(v0, v1) : v0;
#pragma unroll
    for (int o = 16; o > 0; o >>= 1) m = fmaxf(m, __shfl_xor(m, o, 32));
    float ss = expf(v0 - m) + (hi ? expf(v1 - m) : 0.0f);
#pragma unroll
    for (int o = 16; o > 0; o >>= 1) ss += __shfl_xor(ss, o, 32);
    float lse = m + logf(ss);

    float* op = out + (size_t)node * NCLASS;
    op[lane] = v0 - lse;
    if (hi) op[lane + 32] = v1 - lse;
}

// ---------------- launch ----------------

extern "C" void kernel_launch(void* const* d_in, const int* in_sizes, int n_in,
                              void* d_out, int out_size, void* d_ws, size_t ws_size,
                              hipStream_t stream) {
    const float* x    = (const float*)d_in[0];
    const int*   eidx = (const int*)d_in[1];   // jax x64 disabled by default -> int32
    const float* W1   = (const float*)d_in[2];
    const float* b1   = (const float*)d_in[3];
    const float* W2   = (const float*)d_in[4];
    const float* b2   = (const float*)d_in[5];
    const float* W3   = (const float*)d_in[6];
    const float* b3   = (const float*)d_in[7];
    float* out = (float*)d_out;

    const int n = in_sizes[0] / FEAT;   // 50000 (multiple of 16, <= 65536 for scan)
    const int E = in_sizes[1] / 2;      // 800000
    const int* row = eidx;
    const int* col = eidx + E;

    // workspace carve-out (all 4-byte elements; float4 regions stay 16B-aligned)
    char*  base = (char*)d_ws;
    size_t o    = 0;
    auto carve = [&](size_t nelem) { void* p = base + o; o += nelem * 4; return p; };
    float* dinv   = (float*)carve(n);                 // 16B-aligned (n mult of 4)
    float* bufA   = (float*)carve((size_t)n * FEAT);
    float* bufB   = (float*)carve((size_t)n * FEAT);
    float* W3p    = (float*)carve(FEAT * NCLASS_PAD);
    float* normv  = (float*)carve(E);
    int*   cnt    = (int*)carve(n);
    int*   off    = (int*)carve(n + 1);
    int*   cursor = (int*)carve(n);
    int*   srcIdx = (int*)carve(E);
    int*   bsums  = (int*)carve((n + TB - 1) / TB);

    const int gN  = (n + TB - 1) / TB;   // also #scan blocks (196 <= 256)
    const int gE  = (E + TB - 1) / TB;
    const int gNW = ((n * 32) + TB - 1) / TB;   // one wave per node
    const int gMT = n / 16;                     // 3125 M-tiles

    // ---- CSR build (amortized over 3 layers; no float atomics anywhere) ----
    zero_i32<<<gN, TB, 0, stream>>>(cnt, n);
    zero_i32<<<gN, TB, 0, stream>>>(cursor, n);
    count_edges<<<gE, TB, 0, stream>>>(col, cnt, E);
    make_dinv<<<gN, TB, 0, stream>>>(dinv, cnt, n);
    scan_block<<<gN, TB, 0, stream>>>(cnt, off, bsums, n);
    scan_sums<<<1, TB, 0, stream>>>(bsums, gN);
    scan_add<<<gN, TB, 0, stream>>>(off, bsums, n);
    csr_fill<<<gE, TB, 0, stream>>>(row, col, off, cursor, dinv, srcIdx, normv, E);
    pad_w3<<<(FEAT * NCLASS_PAD + TB - 1) / TB, TB, 0, stream>>>(W3, W3p);

    // ----- layer 1 -----
    gemm_wmma_f32<<<gMT, (FEAT / 16) * 32, 0, stream>>>(x, W1, bufA, FEAT);
    gather128<true><<<gNW, TB, 0, stream>>>(bufB, bufA, dinv, b1, off, srcIdx, normv, n);

    // ----- layer 2 -----
    gemm_wmma_f32<<<gMT, (FEAT / 16) * 32, 0, stream>>>(bufB, W2, bufA, FEAT);
    gather128<true><<<gNW, TB, 0, stream>>>(bufB, bufA, dinv, b2, off, srcIdx, normv, n);

    // ----- layer 3 (N padded to 48) + fused log_softmax -----
    gemm_wmma_f32<<<gMT, (NCLASS_PAD / 16) * 32, 0, stream>>>(bufB, W3p, bufA, NCLASS_PAD);
    gather40_lsm<<<gNW, TB, 0, stream>>>(out, bufA, dinv, b3, off, srcIdx, normv, n);

    (void)n_in; (void)out_size; (void)ws_size;
}